// GraphSAGE_83245056131909
// MI455X (gfx1250) — compile-verified
//
#include <hip/hip_runtime.h>

typedef __attribute__((ext_vector_type(2))) float v2f;
typedef __attribute__((ext_vector_type(8))) float v8f;

#define N_NODES 50000
#define N_EDGES 800000
#define D 128
#define N_GRAPHS 512
#define EPS_BN 1e-5f
#define LDS_STRIDE 132   // 128 + 4 pad; row stride 528B = 33*16B keeps 16B alignment

#if __has_builtin(__builtin_amdgcn_global_load_async_to_lds_b128) && \
    __has_builtin(__builtin_amdgcn_s_wait_asynccnt)
#define ASYNC_LDS 1
typedef int v4i __attribute__((vector_size(16)));
typedef __attribute__((address_space(1))) v4i* gvec_t;   // param0: AS1 int4*
typedef __attribute__((address_space(3))) v4i* lvec_t;   // param1: AS3 int4*
#else
#define ASYNC_LDS 0
#endif

// ---------------------------------------------------------------- degree ----
__global__ void deg_kernel(const int* __restrict__ dst, float* __restrict__ deg) {
    int e = blockIdx.x * blockDim.x + threadIdx.x;
    if (e < N_EDGES) atomicAdd(&deg[dst[e]], 1.0f);
}

__global__ void invdeg_kernel(const float* __restrict__ deg, float* __restrict__ inv) {
    int n = blockIdx.x * blockDim.x + threadIdx.x;
    if (n < N_NODES) inv[n] = 1.0f / fmaxf(deg[n], 1.0f);
}

// ------------------------------------------------------------ edge scatter ---
// one lane handles 4 consecutive features of one edge: float4 gather + 4 f32 atomics
__global__ __launch_bounds__(256)
void scatter_kernel(const float* __restrict__ h, const int* __restrict__ src,
                    const int* __restrict__ dst, float* __restrict__ msum) {
    long long gid = (long long)blockIdx.x * blockDim.x + threadIdx.x;
    int e = (int)(gid >> 5);
    if (e >= N_EDGES) return;
    int f = ((int)gid & 31) * 4;
    int s = src[e], d = dst[e];
    float4 v = *(const float4*)(h + (long long)s * D + f);
    float* p = msum + (long long)d * D + f;
    atomicAdd(p + 0, v.x); atomicAdd(p + 1, v.y);
    atomicAdd(p + 2, v.z); atomicAdd(p + 3, v.w);
}

// ---------------------------------------------- fused SAGE + BN + ReLU (WMMA)
// out = relu(bn( inv*(msum@Wl^T) + bl + hin@Wr^T ))   [inv scaling deferred to epilogue]
// block: 16 node-rows x 128 features, 8 waves, each wave one 16x16 tile.
__global__ __launch_bounds__(256)
void sage_gemm_bn_relu(const float* __restrict__ msum, const float* __restrict__ invdeg,
                       const float* __restrict__ hin,
                       const float* __restrict__ Wl, const float* __restrict__ bl,
                       const float* __restrict__ Wr,
                       const float* __restrict__ bnG, const float* __restrict__ bnB,
                       const float* __restrict__ bnM, const float* __restrict__ bnV,
                       float* __restrict__ out) {
    __shared__ __align__(16) float Al[16 * LDS_STRIDE];   // raw msum tile
    __shared__ __align__(16) float Hl[16 * LDS_STRIDE];   // root-feature tile
    __shared__ float IL[16];                              // 1/deg for the 16 rows
    const int row0 = blockIdx.x * 16;
    const int tid  = threadIdx.x;

    // stage 16x128 msum + h tiles into LDS (grid exact: 3125*16 == 50000)
#if ASYNC_LDS
    for (int i = tid; i < 16 * 32; i += 256) {            // 512 x 16B chunks per tile
        int r  = i >> 5;
        int kc = (i & 31) * 4;
        long long goff = (long long)(row0 + r) * D + kc;
        __builtin_amdgcn_global_load_async_to_lds_b128(
            (gvec_t)(msum + goff), (lvec_t)&Al[r * LDS_STRIDE + kc], 0, 0);
        __builtin_amdgcn_global_load_async_to_lds_b128(
            (gvec_t)(hin + goff),  (lvec_t)&Hl[r * LDS_STRIDE + kc], 0, 0);
    }
    if (tid < 16) IL[tid] = invdeg[row0 + tid];
    __builtin_amdgcn_s_wait_asynccnt(0);
    __syncthreads();
#else
    for (int i = tid; i < 16 * D; i += 256) {
        int r = i >> 7, k = i & 127;
        long long goff = (long long)(row0 + r) * D + k;
        Al[r * LDS_STRIDE + k] = msum[goff];
        Hl[r * LDS_STRIDE + k] = hin[goff];
    }
    if (tid < 16) IL[tid] = invdeg[row0 + tid];
    __syncthreads();
#endif

    const int lane = tid & 31;
    const int lo   = lane & 15;        // row (A) / column (B,C,D) within tile
    const int hi   = lane >> 4;        // K-half select (A/B), M-half select (C/D)
    const int n0   = (tid >> 5) * 16;  // this wave's 16-feature column tile
    const int col  = n0 + lo;          // output feature carried by this lane
    const float* wlrow = Wl + (long long)col * D;  // B[k][col] = Wl[col][k]
    const float* wrrow = Wr + (long long)col * D;

    // two independent accumulator chains -> XDL can overlap them
    v8f accL = {}, accR = {};
#pragma unroll
    for (int kb = 0; kb < D; kb += 4) {
        int kk = kb + hi * 2;
        v2f aL = *(const v2f*)(&Al[lo * LDS_STRIDE + kk]);   // A: 16x4 fp32 frag
        v2f bL = *(const v2f*)(wlrow + kk);                  // B: 4x16 fp32 frag
        accL = __builtin_amdgcn_wmma_f32_16x16x4_f32(false, aL, false, bL,
                                                     (short)0, accL, false, false);
        v2f aR = *(const v2f*)(&Hl[lo * LDS_STRIDE + kk]);
        v2f bR = *(const v2f*)(wrrow + kk);
        accR = __builtin_amdgcn_wmma_f32_16x16x4_f32(false, aR, false, bR,
                                                     (short)0, accR, false, false);
    }

    // epilogue: row-scale deferred mean, bias, eval-mode BN, ReLU
    const float bias  = bl[col];
    const float scale = bnG[col] * rsqrtf(bnV[col] + EPS_BN);
    const float shift = bnB[col] - bnM[col] * scale;
#pragma unroll
    for (int i = 0; i < 8; ++i) {
        int m    = i + hi * 8;           // C/D layout: VGPR i -> M=i (+8 for hi lanes)
        int node = row0 + m;
        float v  = (IL[m] * accL[i] + accR[i] + bias) * scale + shift;
        out[(long long)node * D + col] = fmaxf(v, 0.0f);
    }
}

// ------------------------------------------- monotone float<->uint for max ---
__device__ __forceinline__ unsigned enc_f(float f) {
    unsigned u = __float_as_uint(f);
    return (u & 0x80000000u) ? ~u : (u | 0x80000000u);
}
__device__ __forceinline__ float dec_f(unsigned e) {
    return (e & 0x80000000u) ? __uint_as_float(e ^ 0x80000000u) : __uint_as_float(~e);
}

// --------------------------- layer 2 (D_OUT=2) fused with mean/max pooling ---
__global__ __launch_bounds__(256)
void sage_out_pool(const float* __restrict__ msum, const float* __restrict__ invdeg,
                   const float* __restrict__ h1,
                   const float* __restrict__ Wl2, const float* __restrict__ bl2,
                   const float* __restrict__ Wr2, const int* __restrict__ batch,
                   float* __restrict__ psum, float* __restrict__ cnt,
                   unsigned* __restrict__ pmaxenc) {
    int node = blockIdx.x * 8 + (threadIdx.x >> 5);
    int lane = threadIdx.x & 31;
    if (node >= N_NODES) return;
    float inv = invdeg[node];
    float s0 = 0.f, s1 = 0.f;
    for (int f = lane; f < D; f += 32) {
        float a = msum[(long long)node * D + f] * inv;
        float h = h1 [(long long)node * D + f];
        s0 += a * Wl2[f]     + h * Wr2[f];
        s1 += a * Wl2[D + f] + h * Wr2[D + f];
    }
#pragma unroll
    for (int off = 16; off > 0; off >>= 1) {
        s0 += __shfl_xor(s0, off, 32);
        s1 += __shfl_xor(s1, off, 32);
    }
    if (lane == 0) {
        s0 += bl2[0]; s1 += bl2[1];
        int g = batch[node];
        atomicAdd(&psum[g * 2 + 0], s0);
        atomicAdd(&psum[g * 2 + 1], s1);
        atomicAdd(&cnt[g], 1.0f);
        atomicMax(&pmaxenc[g * 2 + 0], enc_f(s0));
        atomicMax(&pmaxenc[g * 2 + 1], enc_f(s1));
    }
}

// -------------------------------------- per-graph MLP: 4 -> 128 -> relu -> 2
__global__ __launch_bounds__(128)
void graph_mlp(const float* __restrict__ psum, const float* __restrict__ cnt,
               const unsigned* __restrict__ pmaxenc,
               const float* __restrict__ mW1, const float* __restrict__ mb1,
               const float* __restrict__ mW2, const float* __restrict__ mb2,
               float* __restrict__ out) {
    __shared__ float r0[128], r1[128];
    int g = blockIdx.x, j = threadIdx.x;
    float c  = fmaxf(cnt[g], 1.0f);
    float g0 = psum[g * 2 + 0] / c, g1 = psum[g * 2 + 1] / c;
    float g2 = dec_f(pmaxenc[g * 2 + 0]), g3 = dec_f(pmaxenc[g * 2 + 1]);
    float hid = mW1[j * 4 + 0] * g0 + mW1[j * 4 + 1] * g1 +
                mW1[j * 4 + 2] * g2 + mW1[j * 4 + 3] * g3 + mb1[j];
    hid = fmaxf(hid, 0.0f);
    r0[j] = mW2[j] * hid;
    r1[j] = mW2[128 + j] * hid;
    __syncthreads();
    for (int s = 64; s > 0; s >>= 1) {
        if (j < s) { r0[j] += r0[j + s]; r1[j] += r1[j + s]; }
        __syncthreads();
    }
    if (j == 0) {
        out[g * 2 + 0] = r0[0] + mb2[0];
        out[g * 2 + 1] = r1[0] + mb2[1];
    }
}

// ---------------------------------------------------------------------------
extern "C" void kernel_launch(void* const* d_in, const int* in_sizes, int n_in,
                              void* d_out, int out_size, void* d_ws, size_t ws_size,
                              hipStream_t stream) {
    (void)in_sizes; (void)n_in; (void)out_size; (void)ws_size;
    const float* x    = (const float*)d_in[0];
    const int*   ei   = (const int*)d_in[1];          // [2, N_EDGES] int32
    const int*   bat  = (const int*)d_in[2];
    const float* Wl0  = (const float*)d_in[3];
    const float* bl0  = (const float*)d_in[4];
    const float* Wr0  = (const float*)d_in[5];
    const float* Wl1  = (const float*)d_in[6];
    const float* bl1  = (const float*)d_in[7];
    const float* Wr1  = (const float*)d_in[8];
    const float* Wl2  = (const float*)d_in[9];
    const float* bl2  = (const float*)d_in[10];
    const float* Wr2  = (const float*)d_in[11];
    const float* g0   = (const float*)d_in[12];
    const float* b0   = (const float*)d_in[13];
    const float* m0   = (const float*)d_in[14];
    const float* v0   = (const float*)d_in[15];
    const float* g1   = (const float*)d_in[16];
    const float* b1   = (const float*)d_in[17];
    const float* m1   = (const float*)d_in[18];
    const float* v1   = (const float*)d_in[19];
    const float* mW1  = (const float*)d_in[20];
    const float* mb1  = (const float*)d_in[21];
    const float* mW2  = (const float*)d_in[22];
    const float* mb2  = (const float*)d_in[23];

    const int* src = ei;
    const int* dst = ei + N_EDGES;

    // workspace carve-up
    char* ws = (char*)d_ws;
    size_t o = 0;
    const size_t NODE_FEAT = (size_t)N_NODES * D * sizeof(float);
    float*    h0     = (float*)(ws + o); o += NODE_FEAT;
    float*    h1     = (float*)(ws + o); o += NODE_FEAT;
    float*    msum   = (float*)(ws + o); o += NODE_FEAT;
    float*    deg    = (float*)(ws + o); o += (size_t)N_NODES * sizeof(float);
    float*    invd   = (float*)(ws + o); o += (size_t)N_NODES * sizeof(float);
    float*    psum   = (float*)(ws + o); o += (size_t)N_GRAPHS * 2 * sizeof(float);
    float*    cnt    = (float*)(ws + o); o += (size_t)N_GRAPHS * sizeof(float);
    unsigned* pmaxe  = (unsigned*)(ws + o);

    const int EB = (N_EDGES + 255) / 256;
    const int NB = (N_NODES + 255) / 256;
    const long long SCAT_T = (long long)N_EDGES * 32;
    const int SB = (int)((SCAT_T + 255) / 256);
    const int GB = N_NODES / 16;          // 3125 exact
    const int PB = (N_NODES + 7) / 8;

    // degree (shared by all layers)
    (void)hipMemsetAsync(deg, 0, (size_t)N_NODES * sizeof(float), stream);
    deg_kernel<<<EB, 256, 0, stream>>>(dst, deg);
    invdeg_kernel<<<NB, 256, 0, stream>>>(deg, invd);

    // layer 0
    (void)hipMemsetAsync(msum, 0, NODE_FEAT, stream);
    scatter_kernel<<<SB, 256, 0, stream>>>(x, src, dst, msum);
    sage_gemm_bn_relu<<<GB, 256, 0, stream>>>(msum, invd, x, Wl0, bl0, Wr0,
                                              g0, b0, m0, v0, h0);
    // layer 1
    (void)hipMemsetAsync(msum, 0, NODE_FEAT, stream);
    scatter_kernel<<<SB, 256, 0, stream>>>(h0, src, dst, msum);
    sage_gemm_bn_relu<<<GB, 256, 0, stream>>>(msum, invd, h0, Wl1, bl1, Wr1,
                                              g1, b1, m1, v1, h1);
    // layer 2 + pooling
    (void)hipMemsetAsync(msum, 0, NODE_FEAT, stream);
    scatter_kernel<<<SB, 256, 0, stream>>>(h1, src, dst, msum);
    (void)hipMemsetAsync(psum, 0, (size_t)N_GRAPHS * 2 * sizeof(float), stream);
    (void)hipMemsetAsync(cnt, 0, (size_t)N_GRAPHS * sizeof(float), stream);
    (void)hipMemsetAsync(pmaxe, 0, (size_t)N_GRAPHS * 2 * sizeof(unsigned), stream);
    sage_out_pool<<<PB, 256, 0, stream>>>(msum, invd, h1, Wl2, bl2, Wr2, bat,
                                          psum, cnt, pmaxe);
    // graph-level MLP -> d_out [512, 2]
    graph_mlp<<<N_GRAPHS, 128, 0, stream>>>(psum, cnt, pmaxe, mW1, mb1, mW2, mb2,
                                            (float*)d_out);
}